// MPIBokehRenderer_22763326669393
// MI455X (gfx1250) — compile-verified
//
#include <hip/hip_runtime.h>
#include <math.h>

#define BINS 32
#define NB   2
#define HH   512
#define WW   512
#define RMAX 40
#define KSZMAX   (2*RMAX+1)          // 81
#define KWSTRIDE (KSZMAX*KSZMAX)     // 6561 floats per bin

// LDS layout (floats), worst case r=RMAX:
//   patch plane: ps * PWpad <= 96 * 148 = 14208
//   wpad       : T * (T+33) <= 81 * 114 = 9234
#define PATCH_F  14208
#define WPAD_OFF (2*PATCH_F)               // two channel planes
#define LDS_F    (WPAD_OFF + 9234)         // 37650 floats
#define LDS_BYTES (LDS_F * 4)              // 150600 B (<320KB/WGP, 2 blocks/WGP)

typedef __attribute__((ext_vector_type(2))) float v2f;
typedef __attribute__((ext_vector_type(8))) float v8f;

// ---------------------------------------------------------------------------
// Kernel 1: per-bin disk kernels + radii on device (scalars are device arrays;
// no host readback under graph capture). Double math matches numpy float64.
// ---------------------------------------------------------------------------
__global__ void prep_kernel(const int* __restrict__ pK,
                            const int* __restrict__ pDisp,
                            float* __restrict__ kw,
                            int*   __restrict__ radii) {
    const int bin = blockIdx.x;
    const int tid = threadIdx.x;          // 64 threads
    const double K    = (double)pK[0];
    const double disp = (double)pDisp[0] / 100.0;
    const double defocus = fabs(((double)bin + 0.5) / (double)BINS - disp);
    int r = (int)rint(defocus * K);       // RNE == python round here
    if (r > RMAX) r = RMAX;
    if (tid == 0) radii[bin] = r;
    const int ks = 2 * r + 1;
    const int n  = ks * ks;
    const double rr = (double)r;

    __shared__ double partial[64];
    __shared__ double total;
    double s = 0.0;
    for (int i = tid; i < n; i += 64) {
        const int dy = i / ks, dx = i % ks;
        const double ddy = (double)dy - rr, ddx = (double)dx - rr;
        double w = rr + 0.5 - sqrt(ddx*ddx + ddy*ddy);
        w = w < 0.0 ? 0.0 : (w > 1.0 ? 1.0 : w);
        s += w;
    }
    partial[tid] = s;
    __syncthreads();
    if (tid == 0) {
        double t = 0.0;
        for (int i = 0; i < 64; ++i) t += partial[i];
        total = t;
    }
    __syncthreads();
    const double inv = 1.0 / total;
    float* kwb = kw + bin * KWSTRIDE;
    for (int i = tid; i < n; i += 64) {
        const int dy = i / ks, dx = i % ks;
        const double ddy = (double)dy - rr, ddx = (double)dx - rr;
        double w = rr + 0.5 - sqrt(ddx*ddx + ddy*ddy);
        w = w < 0.0 ? 0.0 : (w > 1.0 ? 1.0 : w);
        kwb[i] = (float)(w * inv);
    }
}

// ---------------------------------------------------------------------------
// Kernel 2: 4-plane premultiplied source for one bin.
// ---------------------------------------------------------------------------
__global__ void src_kernel(const float* __restrict__ img,
                           const float* __restrict__ alpha,
                           const int*   __restrict__ pGamma,
                           float* __restrict__ src, int bin) {
    const int i = blockIdx.x * 256 + threadIdx.x;   // over NB*H*W
    if (i >= NB * HH * WW) return;
    const int b   = i / (HH * WW);
    const int pix = i % (HH * WW);
    const float gamma = (float)pGamma[0] * 0.1f;
    const float a = alpha[(b * BINS + bin) * (HH * WW) + pix];
    #pragma unroll
    for (int c = 0; c < 3; ++c) {
        const float v = img[(((b * BINS + bin) * 3) + c) * (HH * WW) + pix];
        src[(b * 4 + c) * (HH * WW) + pix] = powf(v, gamma) * a;
    }
    src[(b * 4 + 3) * (HH * WW) + pix] = a;
}

// ---------------------------------------------------------------------------
// Kernel 3: WMMA disk blur + composite for one bin.
// 128 threads = 4 waves; each wave owns one 16x16 tile of a 64-wide strip.
// Banded-Toeplitz GEMM: Out[m,n] = sum_dy sum_k patch[m+dy][wk+k]*Wrow_dy[k-n]
// Weight rows are zero-padded in LDS so B loads are branch-free.
// ---------------------------------------------------------------------------
__global__ __launch_bounds__(128)
void blur_composite_kernel(const float* __restrict__ src,
                           const float* __restrict__ kwAll,
                           const int*   __restrict__ radii,
                           float* __restrict__ bokeh, int bin) {
    extern __shared__ float lds[];
    float* patch0 = lds;                // ps x PWpad, channel pair lo
    float* patch1 = lds + PATCH_F;      // ps x PWpad, channel pair hi
    float* wpad   = lds + WPAD_OFF;     // T rows x (T+33), 15-front/18-back zeros

    const int t    = threadIdx.x;       // 0..127
    const int wid  = t >> 5;            // wave id 0..3
    const int lane = t & 31;
    const int lrow = lane & 15;         // m (A rows) == n (B cols)
    const int klane = (lane >> 4) << 1; // 0 or 2: K sub-offset of this lane

    const int xs = blockIdx.x * 64;     // strip origin
    const int y0 = blockIdx.y * 16;
    const int b  = blockIdx.z;

    const int r     = radii[bin];
    const int T     = 2 * r + 1;
    const int ps    = 16 + 2 * r;                   // per-wave K extent
    const int PWpad = (64 + 2 * r + 2 + 3) & ~3;    // covers col idx <= 65+2r
    const int WSTR  = T + 33;

    // Stage zero-padded weight rows: wpad[dy*WSTR + 15 + dx] = w(dy,dx).
    {
        const float* kwb = kwAll + bin * KWSTRIDE;
        for (int dy = 0; dy < T; ++dy) {
            float* wr = wpad + dy * WSTR;
            if (t < T) wr[15 + t] = kwb[dy * T + t];
            const int pt = t - 96;                   // lanes 96..127 zero pads
            if (pt >= 0 && pt < 33) wr[pt < 15 ? pt : T + pt] = 0.0f;
        }
    }

    v8f acc[4];
    for (int pair = 0; pair < 2; ++pair) {
        const int c0 = 2 * pair;
        // Stage two channel patches with edge clamp (== pad mode='edge').
        const float* sp0 = src + (b * 4 + c0)     * (HH * WW);
        const float* sp1 = src + (b * 4 + c0 + 1) * (HH * WW);
        for (int py = 0; py < ps; ++py) {
            int gy = y0 + py - r; gy = gy < 0 ? 0 : (gy > HH - 1 ? HH - 1 : gy);
            const int rowg = gy * WW;
            const int rowl = py * PWpad;
            for (int px = t; px < PWpad; px += 128) {
                int gx = xs + px - r; gx = gx < 0 ? 0 : (gx > WW - 1 ? WW - 1 : gx);
                patch0[rowl + px] = sp0[rowg + gx];
                patch1[rowl + px] = sp1[rowg + gx];
                if (pair == 0)                        // warm L2/L1 for pair 1
                    __builtin_prefetch(sp0 + 2 * (HH * WW) + rowg + gx, 0, 1);
            }
        }
        __syncthreads();

        v8f a0 = {0.f,0.f,0.f,0.f,0.f,0.f,0.f,0.f};
        v8f a1 = {0.f,0.f,0.f,0.f,0.f,0.f,0.f,0.f};
        for (int dy = 0; dy < T; ++dy) {
            const float* wrow = wpad + dy * WSTR + 15 - lrow;  // wrow[ka]=w[ka-lrow]
            const int rb = (lrow + dy) * PWpad + 16 * wid;
            const float* p0 = patch0 + rb;
            const float* p1 = patch1 + rb;
            #pragma unroll 2
            for (int kk = 0; kk < ps; kk += 4) {
                const int ka = kk + klane;                  // even -> 8B aligned
                const v2f av0 = *(const v2f*)(p0 + ka);     // ds_load_b64
                const v2f av1 = *(const v2f*)(p1 + ka);
                v2f bm;                                     // branch-free B frag
                bm.x = wrow[ka];
                bm.y = wrow[ka + 1];
                a0 = __builtin_amdgcn_wmma_f32_16x16x4_f32(
                        false, av0, false, bm, (short)0, a0, false, false);
                a1 = __builtin_amdgcn_wmma_f32_16x16x4_f32(
                        false, av1, false, bm, (short)0, a1, false, false);
            }
        }
        acc[c0]     = a0;
        acc[c0 + 1] = a1;
        __syncthreads();   // patches reused by next pair
    }

    // Under-composite: bokeh = bokeh*(1 - b_alpha) + b_img   (b_alpha == ch3)
    float* bk = bokeh + b * 4 * (HH * WW);
    const int mhi = 8 * (lane >> 4);
    const int x   = xs + 16 * wid + lrow;
    #pragma unroll
    for (int v = 0; v < 8; ++v) {
        const int y = y0 + v + mhi;
        const int p = y * WW + x;
        const float ba = acc[3][v];
        #pragma unroll
        for (int c = 0; c < 4; ++c) {
            const float blurred = (c == 3) ? ba : acc[c][v];
            const float old = (bin == 0) ? 0.0f : bk[c * (HH * WW) + p];
            bk[c * (HH * WW) + p] = old * (1.0f - ba) + blurred;
        }
    }
}

// ---------------------------------------------------------------------------
// Kernel 4: normalize by accumulated alpha, clip, inverse gamma.
// ---------------------------------------------------------------------------
__global__ void final_kernel(const float* __restrict__ bokeh,
                             const int* __restrict__ pGamma,
                             float* __restrict__ out) {
    const int i = blockIdx.x * 256 + threadIdx.x;   // over NB*3*H*W
    if (i >= NB * 3 * HH * WW) return;
    const int b   = i / (3 * HH * WW);
    const int rem = i % (3 * HH * WW);
    const int c   = rem / (HH * WW);
    const int pix = rem % (HH * WW);
    const float g = (float)pGamma[0] * 0.1f;
    const float num = bokeh[(b * 4 + c) * (HH * WW) + pix];
    const float den = bokeh[(b * 4 + 3) * (HH * WW) + pix];
    float v = num / den;
    v = fminf(fmaxf(v, 1e-10f), 1e10f);
    out[i] = powf(v, 1.0f / g);
}

// ---------------------------------------------------------------------------
extern "C" void kernel_launch(void* const* d_in, const int* in_sizes, int n_in,
                              void* d_out, int out_size, void* d_ws, size_t ws_size,
                              hipStream_t stream) {
    const float* img    = (const float*)d_in[0];  // (2,32,3,512,512)
    const float* alpha  = (const float*)d_in[1];  // (2,32,1,512,512)
    const int*   pK     = (const int*)d_in[2];
    const int*   pDisp  = (const int*)d_in[3];
    const int*   pGamma = (const int*)d_in[4];

    char* ws = (char*)d_ws;
    int*   radii = (int*)ws;                                   // 128 B
    float* kw    = (float*)(ws + 256);                         // ~840 KB
    float* src   = (float*)(ws + (1 << 20));                   // 8 MB
    float* bokeh = (float*)(ws + (1 << 20) + (size_t)NB * 4 * HH * WW * 4);
    float* out   = (float*)d_out;

    prep_kernel<<<BINS, 64, 0, stream>>>(pK, pDisp, kw, radii);

    const int npix = NB * HH * WW;
    const dim3 tgrid(WW / 64, HH / 16, NB);
    for (int bin = 0; bin < BINS; ++bin) {
        src_kernel<<<(npix + 255) / 256, 256, 0, stream>>>(img, alpha, pGamma, src, bin);
        blur_composite_kernel<<<tgrid, 128, LDS_BYTES, stream>>>(src, kw, radii, bokeh, bin);
    }
    const int nout = NB * 3 * HH * WW;
    final_kernel<<<(nout + 255) / 256, 256, 0, stream>>>(bokeh, pGamma, out);
}